// TileUpdate_64802466562424
// MI455X (gfx1250) — compile-verified
//
#include <hip/hip_runtime.h>

// ---------------------------------------------------------------------------
// CDNA5 (gfx1250) implementation of the tile-hypothesis update network.
// Conv layers run on V_WMMA_F32_16X16X4_F32 (full fp32 matrix pipe).
// Warp/cost-volume stage fuses the 3 disparity offsets per thread to cut
// L2 read traffic (fea_l and plane reads shared across the 3 slices).
// ---------------------------------------------------------------------------

typedef __attribute__((ext_vector_type(2))) float v2f;
typedef __attribute__((ext_vector_type(8))) float v8f;

#define DEVI static __device__ __forceinline__

DEVI v8f wmma4(v2f a, v2f b, v8f c) {
  // D = A(16x4) * B(4x16) + C(16x16), all f32. 8-arg VOP3P pattern:
  // (neg_a, A, neg_b, B, c_mod, C, reuse_a, reuse_b)
  return __builtin_amdgcn_wmma_f32_16x16x4_f32(false, a, false, b, (short)0, c,
                                               false, false);
}

DEVI v8f vzero8() {
  v8f z;
#pragma unroll
  for (int i = 0; i < 8; ++i) z[i] = 0.f;
  return z;
}

DEVI float leaky(float x) { return x >= 0.f ? x : 0.2f * x; }
DEVI int iclamp(int v, int lo, int hi) { return v < lo ? lo : (v > hi ? hi : v); }

// Problem dimensions (fixed by the harness).
#define BB 2
#define CF 16
#define HF 384
#define WF 1280
#define HT 96
#define WT 320
#define NPIX (HT * WT)   // 30720
#define NFULL (HF * WF)  // 491520

// ---------------------------------------------------------------------------
// Stage 1: fea = unshuffle_r4( sum_c |fea_l| )  -> (B,16,96,320)
// ---------------------------------------------------------------------------
__global__ void fea_reduce_kernel(const float* __restrict__ fea_l,
                                  float* __restrict__ out) {
  int idx = blockIdx.x * blockDim.x + threadIdx.x;
  if (idx >= BB * NFULL) return;
  int x = idx % WF;
  int y = (idx / WF) % HF;
  int b = idx / NFULL;
  const float* p = fea_l + (size_t)b * CF * NFULL + (size_t)y * WF + x;
  float s = 0.f;
#pragma unroll
  for (int c = 0; c < CF; ++c) s += fabsf(p[(size_t)c * NFULL]);
  int ty = y >> 2, ry = y & 3, tx = x >> 2, rx = x & 3;
  out[(size_t)b * 16 * NPIX + (size_t)(ry * 4 + rx) * NPIX + ty * WT + tx] = s;
}

// ---------------------------------------------------------------------------
// Stage 2: up_prev = upsample_hyp(prev, scale=2, size=2) -> (B,16,96,320)
// ---------------------------------------------------------------------------
__global__ void upsample_hyp_kernel(const float* __restrict__ prev,
                                    float* __restrict__ out) {
  int idx = blockIdx.x * blockDim.x + threadIdx.x;
  if (idx >= BB * 16 * NPIX) return;
  int x = idx % WT;
  int y = (idx / WT) % HT;
  int c = (idx / NPIX) % 16;
  int b = idx / (16 * NPIX);
  const int HP = HT / 2, WP = WT / 2;
  size_t base = (size_t)b * 16 * HP * WP + (size_t)(y >> 1) * WP + (x >> 1);
  float val;
  if (c == 0) {
    float d = prev[base];
    float dx = prev[base + (size_t)1 * HP * WP];
    float dy = prev[base + (size_t)2 * HP * WP];
    float a = (x & 1) ? 0.5f : -0.5f;
    float bcoef = (y & 1) ? 0.5f : -0.5f;
    val = (d + a * dx + bcoef * dy) * 2.0f;
  } else {
    val = prev[base + (size_t)c * HP * WP];
  }
  out[idx] = val;
}

// ---------------------------------------------------------------------------
// Stage 3: tile warping cost volume, all 3 disparity offsets fused per
// thread. For k in {-1,0,1}:
//   local_d(y,x) = d + k + (x%4-1.5)*dx + (y%4-1.5)*dy
//   cv_k = sum_c |fea_l - warp_x(fea_r, local_d)|, unshuffled -> (B,48,96,320)
// One thread per full-res pixel: fea_l and plane reads shared across k.
// ---------------------------------------------------------------------------
__global__ void tile_warp_cv3_kernel(const float* __restrict__ plane,
                                     int plane_ctot,
                                     const float* __restrict__ fea_l,
                                     const float* __restrict__ fea_r,
                                     float* __restrict__ out) {
  int idx = blockIdx.x * blockDim.x + threadIdx.x;
  if (idx >= BB * NFULL) return;
  int x = idx % WF;
  int y = (idx / WF) % HF;
  int b = idx / NFULL;
  int ty = y >> 2, ry = y & 3, tx = x >> 2, rx = x & 3;

  size_t pb = (size_t)b * plane_ctot * NPIX + (size_t)ty * WT + tx;
  float d = plane[pb];
  float dx = plane[pb + NPIX];
  float dy = plane[pb + 2 * NPIX];
  float base_d = d + ((float)rx - 1.5f) * dx + ((float)ry - 1.5f) * dy;

  int xc0[3], xc1[3];
  float w0[3], w1[3];
#pragma unroll
  for (int k = 0; k < 3; ++k) {
    float ld = base_d + (float)(k - 1);
    float ix = (float)x - ld;
    float x0f = floorf(ix);
    float wx = ix - x0f;
    int x0 = (int)x0f;
    int x1 = x0 + 1;
    float ok0 = (x0 >= 0 && x0 < WF) ? 1.f : 0.f;
    float ok1 = (x1 >= 0 && x1 < WF) ? 1.f : 0.f;
    xc0[k] = iclamp(x0, 0, WF - 1);
    xc1[k] = iclamp(x1, 0, WF - 1);
    w0[k] = (1.f - wx) * ok0;
    w1[k] = wx * ok1;
  }

  const float* pl = fea_l + (size_t)b * CF * NFULL + (size_t)y * WF;
  const float* pr = fea_r + (size_t)b * CF * NFULL + (size_t)y * WF;
  float s[3] = {0.f, 0.f, 0.f};
#pragma unroll
  for (int c = 0; c < CF; ++c) {
    float l = pl[(size_t)c * NFULL + x];
    const float* prc = pr + (size_t)c * NFULL;
#pragma unroll
    for (int k = 0; k < 3; ++k) {
      float r = w0[k] * prc[xc0[k]] + w1[k] * prc[xc1[k]];
      s[k] += fabsf(l - r);
    }
  }

  size_t ob = (size_t)b * 48 * NPIX + (size_t)(ry * 4 + rx) * NPIX + ty * WT + tx;
#pragma unroll
  for (int k = 0; k < 3; ++k) out[ob + (size_t)(k * 16) * NPIX] = s[k];
}

// ---------------------------------------------------------------------------
// Stage 4: 1x1 conv (Cin=64 as 4 groups of 16 channels) via WMMA f32 16x16x4.
// One wave -> 16 pixels x NT*16 output channels. Leaky applied on store.
// ---------------------------------------------------------------------------
template <int NT>
__global__ void conv1x1_wmma_kernel(
    const float* __restrict__ g0, int c0t, int c0b,
    const float* __restrict__ g1, int c1t, int c1b,
    const float* __restrict__ g2, int c2t, int c2b,
    const float* __restrict__ g3, int c3t, int c3b,
    const float* __restrict__ Wm,    // (NT*16, 64)
    const float* __restrict__ bias,  // (NT*16)
    float* __restrict__ out, int out_ct) {
  int wave = (blockIdx.x * blockDim.x + threadIdx.x) >> 5;
  int lane = threadIdx.x & 31;
  int half = lane >> 4;
  int lid = lane & 15;
  int b = (wave * 16) / NPIX;
  int r = (wave * 16) % NPIX + lid;

  const float* gp[4] = {g0, g1, g2, g3};
  int gct[4] = {c0t, c1t, c2t, c3t};
  int gcb[4] = {c0b, c1b, c2b, c3b};

  v8f acc[NT];
#pragma unroll
  for (int t = 0; t < NT; ++t) acc[t] = vzero8();

#pragma unroll
  for (int s = 0; s < 16; ++s) {
    int k0 = 4 * s;
    int g = k0 >> 4;
    int cg = (k0 & 15) + 2 * half;
    const float* p =
        gp[g] + (size_t)b * gct[g] * NPIX + (size_t)(gcb[g] + cg) * NPIX + r;
    v2f bv;
    bv.x = p[0];
    bv.y = p[NPIX];
#pragma unroll
    for (int t = 0; t < NT; ++t) {
      int m = lid + 16 * t;
      v2f av;
      av.x = Wm[m * 64 + k0 + 2 * half];
      av.y = Wm[m * 64 + k0 + 2 * half + 1];
      acc[t] = wmma4(av, bv, acc[t]);
    }
  }

#pragma unroll
  for (int t = 0; t < NT; ++t) {
#pragma unroll
    for (int v = 0; v < 8; ++v) {
      int m = 16 * t + 8 * half + v;
      float val = leaky(acc[t][v] + bias[m]);
      out[(size_t)b * out_ct * NPIX + (size_t)m * NPIX + r] = val;
    }
  }
}

// ---------------------------------------------------------------------------
// Stage 5: 3x3 conv (Cin=32, pad=1) via implicit-GEMM WMMA. Weights are
// zero-padded to NT*16 output channels and staged in LDS. Optional residual
// add and leaky. One wave -> 16 consecutive x positions in one row.
// ---------------------------------------------------------------------------
template <int NT>
__global__ void conv3x3_wmma_kernel(const float* __restrict__ in,
                                    const float* __restrict__ Wm,  // (Cout,32,3,3)
                                    const float* __restrict__ bias,
                                    const float* __restrict__ res,
                                    float* __restrict__ out, int Cout,
                                    int leak) {
  const int OP = NT * 16;
  __shared__ float wl[9 * 32 * NT * 16];
  int tid = threadIdx.x;
  for (int i = tid; i < 9 * 32 * OP; i += blockDim.x) {
    int o = i % OP;
    int rest = i / OP;
    int ci = rest % 32;
    int tap = rest / 32;
    wl[i] = (o < Cout) ? Wm[(o * 32 + ci) * 9 + tap] : 0.f;
  }
  __syncthreads();

  int wave = (blockIdx.x * blockDim.x + tid) >> 5;
  int lane = tid & 31;
  int half = lane >> 4;
  int lid = lane & 15;
  const int XT = WT / 16;
  int xt = wave % XT;
  int y = (wave / XT) % HT;
  int b = wave / (XT * HT);
  int x = xt * 16 + lid;

  v8f acc[NT];
#pragma unroll
  for (int t = 0; t < NT; ++t) acc[t] = vzero8();

#pragma unroll
  for (int tap = 0; tap < 9; ++tap) {
    int ky = tap / 3 - 1, kx = tap % 3 - 1;
    int yy = y + ky, xx = x + kx;
    float mask = (yy >= 0 && yy < HT && xx >= 0 && xx < WT) ? 1.f : 0.f;
    int yc = iclamp(yy, 0, HT - 1);
    int xc = iclamp(xx, 0, WT - 1);
    const float* p = in + (size_t)b * 32 * NPIX + (size_t)yc * WT + xc;
#pragma unroll
    for (int s = 0; s < 8; ++s) {
      int c = 4 * s + 2 * half;
      v2f bv;
      bv.x = p[(size_t)c * NPIX] * mask;
      bv.y = p[(size_t)(c + 1) * NPIX] * mask;
#pragma unroll
      for (int t = 0; t < NT; ++t) {
        v2f av;
        av.x = wl[(tap * 32 + c) * OP + 16 * t + lid];
        av.y = wl[(tap * 32 + c + 1) * OP + 16 * t + lid];
        acc[t] = wmma4(av, bv, acc[t]);
      }
    }
  }

  size_t ob = (size_t)b * Cout * NPIX + (size_t)y * WT + x;
#pragma unroll
  for (int t = 0; t < NT; ++t) {
#pragma unroll
    for (int v = 0; v < 8; ++v) {
      int m = 16 * t + 8 * half + v;
      if (m < Cout) {
        float val = acc[t][v] + bias[m];
        if (res) val += res[(size_t)b * 32 * NPIX + (size_t)m * NPIX + (size_t)y * WT + x];
        if (leak) val = leaky(val);
        out[ob + (size_t)m * NPIX] = val;
      }
    }
  }
}

// ---------------------------------------------------------------------------
// Stage 6: split u -> conf/prev_delta/cur_delta, argmax mask, relu-on-d,
// and write the three outputs flat-concatenated into d_out.
// ---------------------------------------------------------------------------
__global__ void finalize_kernel(const float* __restrict__ u,
                                const float* __restrict__ cur,
                                const float* __restrict__ upprev,
                                float* __restrict__ out) {
  int idx = blockIdx.x * blockDim.x + threadIdx.x;
  if (idx >= BB * NPIX) return;
  int b = idx / NPIX;
  int r = idx % NPIX;
  const float* ub = u + (size_t)b * 34 * NPIX + r;
  float conf0 = ub[0];
  float conf1 = ub[(size_t)1 * NPIX];
  float mask = (conf1 > conf0) ? 1.f : 0.f;  // argmax, ties -> index 0

  float* o0 = out;
  float* o1 = out + (size_t)BB * 16 * NPIX;
  float* o2 = o1 + (size_t)BB * 17 * NPIX;

#pragma unroll
  for (int c = 0; c < 16; ++c) {
    float uc = cur[(size_t)b * 16 * NPIX + (size_t)c * NPIX + r] +
               ub[(size_t)(18 + c) * NPIX];
    float up = upprev[(size_t)b * 16 * NPIX + (size_t)c * NPIX + r] +
               ub[(size_t)(2 + c) * NPIX];
    if (c == 0) {
      uc = fmaxf(uc, 0.f);
      up = fmaxf(up, 0.f);
    }
    o0[(size_t)b * 16 * NPIX + (size_t)c * NPIX + r] = mask * uc + (1.f - mask) * up;
    o1[(size_t)b * 17 * NPIX + (size_t)c * NPIX + r] = uc;
    o2[(size_t)b * 17 * NPIX + (size_t)c * NPIX + r] = up;
  }
  o1[(size_t)b * 17 * NPIX + (size_t)16 * NPIX + r] = conf1;
  o2[(size_t)b * 17 * NPIX + (size_t)16 * NPIX + r] = conf0;
}

// ---------------------------------------------------------------------------
extern "C" void kernel_launch(void* const* d_in, const int* in_sizes, int n_in,
                              void* d_out, int out_size, void* d_ws,
                              size_t ws_size, hipStream_t stream) {
  const float* fea_l = (const float*)d_in[0];
  const float* fea_r = (const float*)d_in[1];
  const float* cur = (const float*)d_in[2];
  const float* prev = (const float*)d_in[3];
  const float* w_dec = (const float*)d_in[4];
  const float* b_dec = (const float*)d_in[5];
  const float* w0 = (const float*)d_in[6];
  const float* b0 = (const float*)d_in[7];
  const float* w_r0c1 = (const float*)d_in[8];
  const float* b_r0c1 = (const float*)d_in[9];
  const float* w_r0c2 = (const float*)d_in[10];
  const float* b_r0c2 = (const float*)d_in[11];
  const float* w_r1c1 = (const float*)d_in[12];
  const float* b_r1c1 = (const float*)d_in[13];
  const float* w_r1c2 = (const float*)d_in[14];
  const float* b_r1c2 = (const float*)d_in[15];
  const float* w_last = (const float*)d_in[16];
  const float* b_last = (const float*)d_in[17];

  // Workspace layout (floats). cv is reused for both warps; h aliases cv.
  float* ws = (float*)d_ws;
  float* fea16 = ws;                                 // B*16*NPIX
  float* up_prev = fea16 + (size_t)BB * 16 * NPIX;   // B*16*NPIX
  float* cv = up_prev + (size_t)BB * 16 * NPIX;      // B*48*NPIX
  float* h = cv;                                     // alias: B*32*NPIX (after cv dead)
  float* cur_cv = cv + (size_t)BB * 48 * NPIX;       // B*16*NPIX
  float* prev_cv = cur_cv + (size_t)BB * 16 * NPIX;  // B*16*NPIX
  float* t = prev_cv + (size_t)BB * 16 * NPIX;       // B*32*NPIX
  float* h2 = t + (size_t)BB * 32 * NPIX;            // B*32*NPIX
  float* u = h2 + (size_t)BB * 32 * NPIX;            // B*34*NPIX

  const int waves = BB * NPIX / 16;      // 3840 waves
  const int cblocks = waves * 32 / 128;  // 960 blocks of 4 waves

  {
    int n = BB * NFULL;
    fea_reduce_kernel<<<(n + 255) / 256, 256, 0, stream>>>(fea_l, fea16);
  }
  {
    int n = BB * 16 * NPIX;
    upsample_hyp_kernel<<<(n + 255) / 256, 256, 0, stream>>>(prev, up_prev);
  }
  {
    int n = BB * NFULL;
    tile_warp_cv3_kernel<<<(n + 255) / 256, 256, 0, stream>>>(cur, 16, fea_l,
                                                              fea_r, cv);
  }
  conv1x1_wmma_kernel<1><<<cblocks, 128, 0, stream>>>(
      fea16, 16, 0, cv, 48, 0, cv, 48, 16, cv, 48, 32, w_dec, b_dec, cur_cv, 16);
  {
    int n = BB * NFULL;
    tile_warp_cv3_kernel<<<(n + 255) / 256, 256, 0, stream>>>(up_prev, 16,
                                                              fea_l, fea_r, cv);
  }
  conv1x1_wmma_kernel<1><<<cblocks, 128, 0, stream>>>(
      fea16, 16, 0, cv, 48, 0, cv, 48, 16, cv, 48, 32, w_dec, b_dec, prev_cv, 16);

  // h = leaky(conv1x1(concat[cur, cur_cv, up_prev, prev_cv], w0))  (h aliases cv)
  conv1x1_wmma_kernel<2><<<cblocks, 128, 0, stream>>>(
      cur, 16, 0, cur_cv, 16, 0, up_prev, 16, 0, prev_cv, 16, 0, w0, b0, h, 32);

  conv3x3_wmma_kernel<2><<<cblocks, 128, 0, stream>>>(h, w_r0c1, b_r0c1, nullptr,
                                                      t, 32, 1);
  conv3x3_wmma_kernel<2><<<cblocks, 128, 0, stream>>>(t, w_r0c2, b_r0c2, h, h2,
                                                      32, 1);
  conv3x3_wmma_kernel<2><<<cblocks, 128, 0, stream>>>(h2, w_r1c1, b_r1c1,
                                                      nullptr, t, 32, 1);
  conv3x3_wmma_kernel<2><<<cblocks, 128, 0, stream>>>(t, w_r1c2, b_r1c2, h2, h,
                                                      32, 1);
  conv3x3_wmma_kernel<3><<<cblocks, 128, 0, stream>>>(h, w_last, b_last, nullptr,
                                                      u, 34, 0);
  {
    int n = BB * NPIX;
    finalize_kernel<<<(n + 255) / 256, 256, 0, stream>>>(u, cur, up_prev,
                                                         (float*)d_out);
  }
}